// GCN_45810121179684
// MI455X (gfx1250) — compile-verified
//
#include <hip/hip_runtime.h>

// GCN (3x GraphConv) for MI455X / gfx1250.
// Memory-bound on edge gather/scatter (~5GB -> ~0.25ms @ 23.3TB/s); GEMMs (~10 GFLOP)
// done with V_WMMA_F32_16X16X4_F32 to keep full fp32 precision while using the matrix pipe.
// Scatter-add uses hardware global_atomic_add_f32 via unsafeAtomicAdd (no CAS loop).

#define NNODES 100000
#define NEDGES 1600000
#define DIM    128
#define NCLS   40

typedef __attribute__((ext_vector_type(2))) float v2f;
typedef __attribute__((ext_vector_type(8))) float v8f;

// Hardware float atomic add (coarse-grained HBM workspace -> safe).
__device__ __forceinline__ void atomic_fadd_hw(float* p, float v) {
    unsafeAtomicAdd(p, v);
}

// ---------------- utility kernels ----------------

__global__ __launch_bounds__(256) void zero_f4(float4* __restrict__ p, int n4) {
    int i = blockIdx.x * 256 + threadIdx.x;
    if (i < n4) p[i] = make_float4(0.f, 0.f, 0.f, 0.f);
}

__global__ __launch_bounds__(256) void degree_kernel(const int* __restrict__ src,
                                                     const int* __restrict__ dst,
                                                     float* __restrict__ degs,
                                                     float* __restrict__ degd) {
    int e = blockIdx.x * 256 + threadIdx.x;
    if (e < NEDGES) {
        atomic_fadd_hw(degs + src[e], 1.0f);
        atomic_fadd_hw(degd + dst[e], 1.0f);
    }
}

__global__ __launch_bounds__(256) void norm_finalize(float* __restrict__ d, int n) {
    int i = blockIdx.x * 256 + threadIdx.x;
    if (i < n) {
        float v = d[i];
        d[i] = (v > 0.f) ? rsqrtf(fmaxf(v, 1.f)) : 0.f;
    }
}

// ---------------- edge scatter: agg[dst] += h[src] * norm_src[src] ----------------
// One edge per wave32: lane L moves 4 contiguous floats (float4 load, 4 atomic fadds).
// Row loads are fully coalesced (512B/wave); src/dst/ns loads are wave-uniform broadcasts.

__global__ __launch_bounds__(256) void scatter_kernel(const float* __restrict__ h,
                                                      const float* __restrict__ ns,
                                                      const int* __restrict__ src,
                                                      const int* __restrict__ dst,
                                                      float* __restrict__ agg) {
    int e    = (blockIdx.x * 256 + threadIdx.x) >> 5;  // one edge per wave
    int lane = threadIdx.x & 31;
    int s = src[e];
    int d = dst[e];
    float sc = ns[s];
    float4 v = ((const float4*)(h + (size_t)s * DIM))[lane];
    float* base = agg + (size_t)d * DIM + lane * 4;
    atomic_fadd_hw(base + 0, v.x * sc);
    atomic_fadd_hw(base + 1, v.y * sc);
    atomic_fadd_hw(base + 2, v.z * sc);
    atomic_fadd_hw(base + 3, v.w * sc);
}

// ---------------- WMMA GEMM: out = act( (agg * norm_dst) @ W + b ) ----------------
// Block = 16 nodes (one M-tile) x 8 waves; wave w computes N-tile [16w, 16w+16).
// A tile staged in LDS (padded stride 132 -> conflict-free column reads),
// W staged in LDS (zero-padded to NT*16 columns for the 40-class layer).
// K=128 consumed as 32 chained v_wmma_f32_16x16x4_f32 ops.

template <int NOUT, int NT, bool RELU>
__global__ __launch_bounds__(256) void gcn_gemm(const float* __restrict__ agg,
                                                const float* __restrict__ nd,
                                                const float* __restrict__ W,
                                                const float* __restrict__ bias,
                                                float* __restrict__ out) {
    constexpr int NT16 = NT * 16;
    __shared__ float As[16][DIM + 4];
    __shared__ float Ws[DIM][NT16];

    const int t    = threadIdx.x;
    const int tile = blockIdx.x;  // 100000/16 = 6250 exact

    // Stage A (scaled by norm_dst): 16 threads per row, 8 floats each.
    {
        int m    = t >> 4;          // 0..15
        int kb   = (t & 15) * 8;    // 0..120
        int node = tile * 16 + m;
        float s  = nd[node];
        const float4* p = (const float4*)(agg + (size_t)node * DIM + kb);
        float4 v0 = p[0], v1 = p[1];
        As[m][kb + 0] = v0.x * s; As[m][kb + 1] = v0.y * s;
        As[m][kb + 2] = v0.z * s; As[m][kb + 3] = v0.w * s;
        As[m][kb + 4] = v1.x * s; As[m][kb + 5] = v1.y * s;
        As[m][kb + 6] = v1.z * s; As[m][kb + 7] = v1.w * s;
    }
    // Stage W (zero-fill columns >= NOUT).
    for (int i = t; i < DIM * NT16; i += 256) {
        int k = i / NT16;
        int n = i - k * NT16;
        Ws[k][n] = (n < NOUT) ? W[k * NOUT + n] : 0.0f;
    }
    __syncthreads();

    const int wave = t >> 5;
    const int lane = t & 31;
    if (wave < NT) {                 // wave-uniform: EXEC all-1s inside (WMMA requirement)
        const int n0  = wave * 16;
        const int l16 = lane & 15;
        const int kh  = (lane >> 4) * 2;   // ISA A/B layout: K-half selected by lane half
        v8f acc = {};
#pragma unroll
        for (int kk = 0; kk < DIM; kk += 4) {
            v2f a, b;
            a.x = As[l16][kk + kh];          // A: lane = row M, vgpr = K within half
            a.y = As[l16][kk + kh + 1];
            b.x = Ws[kk + kh][n0 + l16];     // B: lane = col N, vgpr = K within half
            b.y = Ws[kk + kh + 1][n0 + l16];
            acc = __builtin_amdgcn_wmma_f32_16x16x4_f32(
                false, a, false, b, (short)0, acc, false, false);
        }
        // D layout: vgpr v, lane half h -> M = v + 8h, N = lane&15.
        const int n = n0 + l16;
        if (n < NOUT) {
            const int mbase = tile * 16 + (lane >> 4) * 8;
            float bv = bias[n];
#pragma unroll
            for (int v = 0; v < 8; ++v) {
                float r = acc[v] + bv;
                if (RELU) r = fmaxf(r, 0.0f);
                out[(size_t)(mbase + v) * NOUT + n] = r;
            }
        }
    }
}

// ---------------- host side ----------------

extern "C" void kernel_launch(void* const* d_in, const int* in_sizes, int n_in,
                              void* d_out, int out_size, void* d_ws, size_t ws_size,
                              hipStream_t stream) {
    const float* inputs = (const float*)d_in[0];
    const int*   src    = (const int*)d_in[1];
    const int*   dst    = (const int*)d_in[2];
    const float* W0     = (const float*)d_in[3];
    const float* b0     = (const float*)d_in[4];
    const float* W1     = (const float*)d_in[5];
    const float* b1     = (const float*)d_in[6];
    const float* W2     = (const float*)d_in[7];
    const float* b2     = (const float*)d_in[8];
    float* out = (float*)d_out;

    float* ws   = (float*)d_ws;
    float* ns   = ws;                       // norm_src [N]
    float* ndm  = ws + NNODES;              // norm_dst [N]
    float* bufB = ws + 2 * NNODES;          // agg      [N*128]
    float* bufC = bufB + (size_t)NNODES * DIM;  // hidden  [N*128]

    constexpr int FEAT4   = NNODES * DIM / 4;         // 3.2M float4
    constexpr int GR_FEAT = FEAT4 / 256;              // 12500
    constexpr int NORM4   = 2 * NNODES / 4;           // 50000 float4
    constexpr int GR_NZ   = (NORM4 + 255) / 256;      // 196
    constexpr int GR_DEG  = (NEDGES + 255) / 256;     // 6250
    constexpr int GR_NF   = (2 * NNODES + 255) / 256; // 782
    constexpr int GR_SCAT = NEDGES / 8;               // 200000 (8 edges/block)
    constexpr int GR_GEMM = NNODES / 16;              // 6250

    // Degree norms (reference: rsqrt(max(deg,1)) where deg>0 else 0).
    zero_f4<<<GR_NZ, 256, 0, stream>>>((float4*)ns, NORM4);
    degree_kernel<<<GR_DEG, 256, 0, stream>>>(src, dst, ns, ndm);
    norm_finalize<<<GR_NF, 256, 0, stream>>>(ns, 2 * NNODES);

    // Layer 0: inputs -> bufC
    zero_f4<<<GR_FEAT, 256, 0, stream>>>((float4*)bufB, FEAT4);
    scatter_kernel<<<GR_SCAT, 256, 0, stream>>>(inputs, ns, src, dst, bufB);
    gcn_gemm<DIM, 8, true><<<GR_GEMM, 256, 0, stream>>>(bufB, ndm, W0, b0, bufC);

    // Layer 1: bufC -> bufC (agg in bufB; bufC free once scatter completes)
    zero_f4<<<GR_FEAT, 256, 0, stream>>>((float4*)bufB, FEAT4);
    scatter_kernel<<<GR_SCAT, 256, 0, stream>>>(bufC, ns, src, dst, bufB);
    gcn_gemm<DIM, 8, true><<<GR_GEMM, 256, 0, stream>>>(bufB, ndm, W1, b1, bufC);

    // Layer 2: bufC -> logits (no ReLU), 40 classes -> 3 N-tiles
    zero_f4<<<GR_FEAT, 256, 0, stream>>>((float4*)bufB, FEAT4);
    scatter_kernel<<<GR_SCAT, 256, 0, stream>>>(bufC, ns, src, dst, bufB);
    gcn_gemm<NCLS, 3, false><<<GR_GEMM, 256, 0, stream>>>(bufB, ndm, W2, b2, out);
}